// HomoHingeLoss_55207509623499
// MI455X (gfx1250) — compile-verified
//
#include <hip/hip_runtime.h>

typedef __attribute__((ext_vector_type(2))) float v2f;
typedef __attribute__((ext_vector_type(8))) float v8f;
typedef unsigned int u32;
typedef __attribute__((ext_vector_type(4))) u32 v4u;
typedef __attribute__((ext_vector_type(8))) int v8i;
typedef __attribute__((ext_vector_type(4))) int v4i;

#define GRID_SZ    8
#define POS_LAMBDA 250.0f
#define POS_MARGIN 1.0f
#define NEG_MARGIN 0.2f
#define NB  2
#define CC  256
#define HH  48
#define WW  64
#define PP  (HH * WW)            /* 3072 */

#define BPT 64                   /* block P-tile (2 waves x 32)  */
#define BQT 128                  /* block Q-tile (4 waves x 32)  */
#define KC  32                   /* K chunk staged in LDS        */
#define ASTRIDE 72               /* 64 + TDM pad 8: 2*72 mod 64 = 16 -> conflict-free */
#define BSTRIDE 136              /* 128 + TDM pad 8: 2*136 mod 64 = 16 -> conflict-free */
#define ABUF (KC * ASTRIDE)      /* 2304 floats */
#define BBUF (KC * BSTRIDE)      /* 4352 floats */
#define BUFSZ (ABUF + BBUF)      /* 6656 floats per buffer */
#define NPARTS ((PP / BPT) * (PP / BQT) * NB)   /* 48*24*2 = 2304 */

/* ---------- CDNA5 Tensor Data Mover: 2-D tile Global -> LDS ----------
 * D# group0: count=1 | lds_addr | 57b global_addr | type=2
 * D# group1: data_size=4B, pad cfg, tensor_dim0/1, tile_dim0/1,
 *            tensor_dim0_stride (all in element units)
 * Groups 2/3 zero (2-D tensor). Tracked by TENSORcnt.                  */
__device__ __forceinline__ void tdm_load_2d(u32 lds_byte_off, const float* gptr,
                                            u32 tile_d0, u32 tile_d1,
                                            u32 tensor_d0, u32 tensor_d1,
                                            u32 stride0_elems,
                                            u32 pad_interval_code, u32 pad_amount_code) {
    const unsigned long long ga = (unsigned long long)(size_t)(const void*)gptr;
    v4u g0;
    g0[0] = 1u;                                        /* count=1, user mode   */
    g0[1] = lds_byte_off;                              /* lds_addr (bytes)     */
    g0[2] = (u32)ga;                                   /* global_addr[31:0]    */
    g0[3] = (u32)((ga >> 32) & 0x1FFFFFFu) | (2u << 30); /* addr[56:32]|type=2 */
    v8i g1;
    g1[0] = (int)((2u << 16) |                         /* data_size: 4 bytes   */
                  (1u << 20) |                         /* pad_enable           */
                  (pad_interval_code << 22) |
                  (pad_amount_code << 25));
    g1[1] = (int)((tensor_d0 & 0xFFFFu) << 16);        /* tensor_dim0[15:0]    */
    g1[2] = (int)((tensor_d0 >> 16) |
                  ((tensor_d1 & 0xFFFFu) << 16));      /* d0[31:16] | d1[15:0] */
    g1[3] = (int)((tensor_d1 >> 16) | (tile_d0 << 16));/* d1[31:16] | tile_d0  */
    g1[4] = (int)tile_d1;                              /* tile_dim1, tile_dim2=0 */
    g1[5] = (int)stride0_elems;                        /* tensor_dim0_stride   */
    g1[6] = 0;
    g1[7] = 0;
    const v4i z4 = {0, 0, 0, 0};
    const v8i z8 = {0, 0, 0, 0, 0, 0, 0, 0};
    /* 6-arg variant (clang-23 / therock-10.0 headers) */
    __builtin_amdgcn_tensor_load_to_lds(g0, g1, z4, z4, z8, 0);
}

__device__ __forceinline__ u32 lds_off(const void* p) {
    /* generic LDS pointer: low 32 bits are the LDS byte offset (ISA 10.2) */
    return (u32)(size_t)p;
}

__device__ __forceinline__ float hinge_tile(const v8f& acc, int pTile,
                                            float wx, float wy, int hs) {
    float s = 0.f;
    #pragma unroll
    for (int v = 0; v < 8; ++v) {
        const int   p   = pTile + v + hs * 8;      /* C-layout row */
        const float gx1 = (float)((p % WW) * GRID_SZ + GRID_SZ / 2);
        const float gy1 = (float)((p / WW) * GRID_SZ + GRID_SZ / 2);
        const float dx  = gx1 - wx, dy = gy1 - wy;
        const bool  m   = (dx * dx + dy * dy) <=
                          (GRID_SZ - 0.5f) * (GRID_SZ - 0.5f);
        const float dot = acc[v];
        const float pos = fmaxf(POS_MARGIN - dot, 0.f);
        const float neg = fmaxf(dot - NEG_MARGIN, 0.f);
        s += m ? POS_LAMBDA * pos : neg;
    }
    return s;
}

#define WMMA_F32(A, B, Cacc) \
    __builtin_amdgcn_wmma_f32_16x16x4_f32(false, (A), false, (B), (short)0, (Cacc), false, false)

__global__ __launch_bounds__(256)
void HomoHinge_gemm_kernel(const float* __restrict__ desc1,
                           const float* __restrict__ desc2,
                           const float* __restrict__ homo,
                           float* __restrict__ partial) {
    __shared__ float smem[2 * BUFSZ];              /* 52 KB double buffer */

    const int qBlk  = blockIdx.x;                  /* 0..23 */
    const int pBlk  = blockIdx.y;                  /* 0..47 */
    const int nb    = blockIdx.z;                  /* 0..1  */
    const int pBase = pBlk * BPT;
    const int qBase = qBlk * BQT;

    const int t    = threadIdx.x;
    const int lane = t & 31;
    const int wave = t >> 5;
    const int wp   = wave & 1;                     /* 2 p-waves of 32 rows */
    const int wq   = wave >> 1;                    /* 4 q-waves of 32 cols */
    const int mn   = lane & 15;
    const int hs   = lane >> 4;

    const float* Ag = desc1 + (size_t)nb * CC * PP;
    const float* Bg = desc2 + (size_t)nb * CC * PP;

    /* homography (uniform, cache-resident) */
    const float h00 = homo[0], h01 = homo[1], h02 = homo[2];
    const float h10 = homo[3], h11 = homo[4], h12 = homo[5];
    const float h20 = homo[6], h21 = homo[7], h22 = homo[8];

    /* this lane's two output columns (fixed across all acc rows) */
    const int q0 = qBase + wq * 32 + mn;
    float wxs[2], wys[2];
    #pragma unroll
    for (int sq = 0; sq < 2; ++sq) {
        const int   q   = q0 + sq * 16;
        const float gx2 = (float)((q % WW) * GRID_SZ + GRID_SZ / 2);
        const float gy2 = (float)((q / WW) * GRID_SZ + GRID_SZ / 2);
        const float inv = 1.0f / (h20 * gx2 + h21 * gy2 + h22);
        wxs[sq] = (h00 * gx2 + h01 * gy2 + h02) * inv;
        wys[sq] = (h10 * gx2 + h11 * gy2 + h12) * inv;
    }

    v8f acc00 = {0.f,0.f,0.f,0.f,0.f,0.f,0.f,0.f};
    v8f acc01 = acc00, acc10 = acc00, acc11 = acc00;

    /* TDM software pipeline: wave0 DMAs the A tile, wave1 the B tile.
     * Pad codes: A rows 64 dw -> pad 8 every 64 (interval 5, amount 7) => stride 72
     *            B rows 128 dw -> pad 8 every 128 (interval 6, amount 7) => stride 136 */
    if (wave == 0)
        tdm_load_2d(lds_off(smem), Ag + pBase,
                    BPT, KC, PP, CC, PP, 5u, 7u);
    else if (wave == 1)
        tdm_load_2d(lds_off(smem + ABUF), Bg + qBase,
                    BQT, KC, PP, CC, PP, 6u, 7u);

    for (int kc = 0; kc < CC; kc += KC) {
        if (wave < 2) __builtin_amdgcn_s_wait_tensorcnt(0);
        __syncthreads();                            /* chunk kc resident */
        const int    buf = (kc / KC) & 1;
        const float* As  = smem + buf * BUFSZ;
        const float* Bs  = As + ABUF;

        /* overlap: TDM-stage chunk kc+KC into the other buffer */
        if (kc + KC < CC) {
            const int nbuf = buf ^ 1;
            if (wave == 0)
                tdm_load_2d(lds_off(smem + nbuf * BUFSZ),
                            Ag + (size_t)(kc + KC) * PP + pBase,
                            BPT, KC, PP, CC, PP, 5u, 7u);
            else if (wave == 1)
                tdm_load_2d(lds_off(smem + nbuf * BUFSZ + ABUF),
                            Bg + (size_t)(kc + KC) * PP + qBase,
                            BQT, KC, PP, CC, PP, 6u, 7u);
        }

        /* f32 WMMA, 2x2 register blocking: 4 ds-pair loads feed 4 wmma */
        #pragma unroll 4
        for (int ks = 0; ks < KC / 4; ++ks) {
            const int k0 = ks * 4 + hs * 2;
            const float* Ab = As + k0 * ASTRIDE + wp * 32 + mn;
            const float* Bb = Bs + k0 * BSTRIDE + wq * 32 + mn;
            v2f a0, a1, b0, b1;
            a0.x = Ab[0];   a0.y = Ab[ASTRIDE];
            a1.x = Ab[16];  a1.y = Ab[ASTRIDE + 16];
            b0.x = Bb[0];   b0.y = Bb[BSTRIDE];
            b1.x = Bb[16];  b1.y = Bb[BSTRIDE + 16];
            acc00 = WMMA_F32(a0, b0, acc00);
            acc01 = WMMA_F32(a0, b1, acc01);
            acc10 = WMMA_F32(a1, b0, acc10);
            acc11 = WMMA_F32(a1, b1, acc11);
        }
    }

    /* fused epilogue: hinge + mask on 4 register tiles */
    const int pT0 = pBase + wp * 32;
    float lsum = 0.f;
    lsum += hinge_tile(acc00, pT0,      wxs[0], wys[0], hs);
    lsum += hinge_tile(acc01, pT0,      wxs[1], wys[1], hs);
    lsum += hinge_tile(acc10, pT0 + 16, wxs[0], wys[0], hs);
    lsum += hinge_tile(acc11, pT0 + 16, wxs[1], wys[1], hs);

    /* deterministic block tree-reduction (reuse LDS) */
    __syncthreads();
    smem[t] = lsum;
    __syncthreads();
    for (int off = 128; off > 0; off >>= 1) {
        if (t < off) smem[t] += smem[t + off];
        __syncthreads();
    }
    if (t == 0)
        partial[(nb * gridDim.y + pBlk) * gridDim.x + qBlk] = smem[0];
}

__global__ __launch_bounds__(256)
void HomoHinge_reduce_kernel(const float* __restrict__ partial,
                             float* __restrict__ out,
                             int nParts, float invCount) {
    __shared__ float smem[256];
    float s = 0.f;
    for (int i = threadIdx.x; i < nParts; i += 256) s += partial[i];
    smem[threadIdx.x] = s;
    __syncthreads();
    for (int off = 128; off > 0; off >>= 1) {
        if ((int)threadIdx.x < off) smem[threadIdx.x] += smem[threadIdx.x + off];
        __syncthreads();
    }
    if (threadIdx.x == 0) out[0] = smem[0] * invCount;
}

extern "C" void kernel_launch(void* const* d_in, const int* in_sizes, int n_in,
                              void* d_out, int out_size, void* d_ws, size_t ws_size,
                              hipStream_t stream) {
    (void)in_sizes; (void)n_in; (void)out_size; (void)ws_size;
    const float* desc1 = (const float*)d_in[0];
    const float* desc2 = (const float*)d_in[1];
    const float* homo  = (const float*)d_in[2];
    float* out = (float*)d_out;
    float* ws  = (float*)d_ws;             /* NPARTS floats = 9 KB */

    dim3 grid(PP / BQT, PP / BPT, NB);     /* (24, 48, 2) = 2304 blocks */
    HomoHinge_gemm_kernel<<<grid, 256, 0, stream>>>(desc1, desc2, homo, ws);

    const float invCount = 1.0f / ((float)NB * (float)PP * (float)PP);
    HomoHinge_reduce_kernel<<<1, 256, 0, stream>>>(ws, out, NPARTS, invCount);
}